// FunBERT_42872363549008
// MI455X (gfx1250) — compile-verified
//
#include <hip/hip_runtime.h>
#include <math.h>

// Problem constants (from reference): B=64, S=512, D=1536
#define Bn 64
#define Sn 512
#define Dn 1536
#define FD (4 * Dn)   // 6144 feature dim
#define Hn 1024       // hidden dim

typedef __attribute__((ext_vector_type(2))) float v2f;
typedef __attribute__((ext_vector_type(8))) float v8f;

#if defined(__gfx1250__) && __has_builtin(__builtin_amdgcn_wmma_f32_16x16x4_f32)
#define HAVE_WMMA_F32X4 1
#else
#define HAVE_WMMA_F32X4 0
#endif

// ---------------------------------------------------------------------------
// Kernel A: masked means over tiny windows -> combined entity matrix E
// E rows 0..63   = entity2[b]  (feeds q2 / pool p=0, and feat slot 3)
// E rows 64..127 = entity1[b]  (feeds q1 / pool p=1)
// ---------------------------------------------------------------------------
__global__ void entities_kernel(const float* __restrict__ x,
                                const int* __restrict__ locs,
                                float* __restrict__ E) {
    int b = blockIdx.x;
    int l0 = locs[b * 4 + 0], l1 = locs[b * 4 + 1];
    int l2 = locs[b * 4 + 2], l3 = locs[b * 4 + 3];
    float inv1 = 1.0f / (float)(l1 - l0 - 1);   // window sizes guaranteed >= 1
    float inv2 = 1.0f / (float)(l3 - l2 - 1);
    const float* xb = x + (size_t)b * Sn * Dn;
    for (int d = threadIdx.x; d < Dn; d += blockDim.x) {
        float s1 = 0.f, s2 = 0.f;
        for (int s = l0 + 1; s < l1; ++s) s1 += xb[(size_t)s * Dn + d];
        for (int s = l2 + 1; s < l3; ++s) s2 += xb[(size_t)s * Dn + d];
        E[(size_t)b * Dn + d]        = s2 * inv2;   // entity2 -> rows 0..63
        E[(size_t)(64 + b) * Dn + d] = s1 * inv1;   // entity1 -> rows 64..127
    }
}

// ---------------------------------------------------------------------------
// WMMA GEMM: Dst[m][n] = sum_k A[m][k] * W[n][k]   (A row-major MxK, W NxK)
// One wave computes one 16x16 tile using V_WMMA_F32_16X16X4_F32.
// fp32 A-frag (16x4): lane L<16 -> row L, K pair {k0,k0+1}; lane>=16 -> {k0+2,k0+3}
// fp32 B-frag (4x16): symmetric: lane carries column n=lane&15, same K pair.
// C/D frag: vgpr r -> m = 16*tm + r + 8*(lane>=16), n = 16*tn + (lane&15)
// EPILOGUE: +bias[n], leaky-relu with *alphap
// ---------------------------------------------------------------------------
template <bool EPILOGUE>
__global__ void wmma_gemm_kernel(const float* __restrict__ A,
                                 const float* __restrict__ W,
                                 float* __restrict__ Dst,
                                 int Mtiles, int Ntiles, int K, int N,
                                 const float* __restrict__ bias,
                                 const float* __restrict__ alphap) {
    int wave = threadIdx.x >> 5;
    int lane = threadIdx.x & 31;
    int tile = blockIdx.x * (blockDim.x >> 5) + wave;
    int tm = tile % Mtiles;
    int tn = tile / Mtiles;
    if (tn >= Ntiles) return;

    int l15   = lane & 15;
    int lhalf = lane >> 4;          // 0: K pair {0,1}, 1: K pair {2,3}
    int n     = tn * 16 + l15;
    int mbase = tm * 16 + 8 * lhalf;

    const float* arow = A + (size_t)(tm * 16 + l15) * K;  // A row for this lane's frag
    const float* wrow = W + (size_t)n * K;                // W row n (B-frag source)

    v8f acc = {0.f, 0.f, 0.f, 0.f, 0.f, 0.f, 0.f, 0.f};

    for (int k0 = 0; k0 < K; k0 += 4) {
#if HAVE_WMMA_F32X4
        v2f a = *(const v2f*)(arow + k0 + 2 * lhalf);
        v2f b = *(const v2f*)(wrow + k0 + 2 * lhalf);
        acc = __builtin_amdgcn_wmma_f32_16x16x4_f32(
            /*neg_a=*/false, a, /*neg_b=*/false, b,
            /*c_mod=*/(short)0, acc, /*reuse_a=*/false, /*reuse_b=*/false);
#else
        // scalar fallback (compile safety only)
        for (int r = 0; r < 8; ++r) {
            const float* ar = A + (size_t)(mbase + r) * K;
            float s = acc[r];
            for (int kk = 0; kk < 4; ++kk) s += ar[k0 + kk] * wrow[k0 + kk];
            acc[r] = s;
        }
#endif
    }

    if (EPILOGUE) {
        float bv = bias[n];
        float al = alphap[0];
        for (int r = 0; r < 8; ++r) {
            float h = acc[r] + bv;
            h = (h >= 0.f) ? h : al * h;
            Dst[(size_t)(mbase + r) * N + n] = h;
        }
    } else {
        for (int r = 0; r < 8; ++r)
            Dst[(size_t)(mbase + r) * N + n] = acc[r];
    }
}

// ---------------------------------------------------------------------------
// Kernel C: masked-softmax attention pool. One block per (b, p).
//   p=0: imp2 mask, q row b      -> P row b        (sent_attn2)
//   p=1: imp1 mask, q row 64+b   -> P row 64+b     (sent_attn1)
// Phase 1: wave-per-s dot products (lane-parallel over D, coalesced).
// Phase 2: block softmax in LDS (two-pass: max, then exp+sum over 512 scores).
// Phase 3: thread-parallel over D weighted accumulation (coalesced).
// ---------------------------------------------------------------------------
__global__ void attn_pool_kernel(const float* __restrict__ x,
                                 const float* __restrict__ Q,
                                 const int* __restrict__ locs,
                                 const int* __restrict__ seps,
                                 float* __restrict__ P) {
    __shared__ float sc[Sn];
    __shared__ float redmax[8];
    __shared__ float redsum[8];

    int b = blockIdx.x;
    int p = blockIdx.y;
    int tid  = threadIdx.x;          // 256 threads = 8 wave32
    int wave = tid >> 5;
    int lane = tid & 31;

    int l0 = locs[b * 4 + 0], l1 = locs[b * 4 + 1];
    int l2 = locs[b * 4 + 2], l3 = locs[b * 4 + 3];
    int sep0 = seps[b * 2 + 0], sep1 = seps[b * 2 + 1];

    const float* q  = Q + (size_t)(p * 64 + b) * Dn;
    const float* xb = x + (size_t)b * Sn * Dn;

    // Phase 1: scores[s] = <q, x[b,s,:]> where masked, else -inf
    for (int s = wave; s < Sn; s += 8) {
        bool m;
        if (p == 0) m = ((s >= sep1) && (s <= l2)) || (s >= l3);        // imp2
        else        m = (s <= l0) || ((s >= l1) && (s < sep0));         // imp1
        float partial = 0.f;
        if (m) {
            const float* xs = xb + (size_t)s * Dn;
            for (int d = lane; d < Dn; d += 32) partial += q[d] * xs[d];
            for (int off = 16; off > 0; off >>= 1)
                partial += __shfl_xor(partial, off, 32);
        }
        if (lane == 0) sc[s] = m ? partial : -__builtin_inff();
    }
    __syncthreads();

    // Phase 2a: block max
    float mymax = -__builtin_inff();
    for (int s = tid; s < Sn; s += 256) mymax = fmaxf(mymax, sc[s]);
    for (int off = 16; off > 0; off >>= 1)
        mymax = fmaxf(mymax, __shfl_xor(mymax, off, 32));
    if (lane == 0) redmax[wave] = mymax;
    __syncthreads();
    float gmax = redmax[0];
    for (int w = 1; w < 8; ++w) gmax = fmaxf(gmax, redmax[w]);

    // Phase 2b: exp + sum (each thread only touches its own sc[] entries)
    float mysum = 0.f;
    for (int s = tid; s < Sn; s += 256) {
        float e = __expf(sc[s] - gmax);   // exp(-inf) -> 0 for masked slots
        sc[s] = e;
        mysum += e;
    }
    for (int off = 16; off > 0; off >>= 1)
        mysum += __shfl_xor(mysum, off, 32);
    if (lane == 0) redsum[wave] = mysum;
    __syncthreads();
    float gsum = 0.f;
    for (int w = 0; w < 8; ++w) gsum += redsum[w];
    float inv = 1.0f / gsum;

    // Phase 3: pooled[d] = inv * sum_s sc[s] * x[b,s,d]
    for (int d = tid; d < Dn; d += 256) {
        float acc = 0.f;
        for (int s = 0; s < Sn; ++s) {
            if ((s & 7) == 0 && s + 8 < Sn)
                __builtin_prefetch(xb + (size_t)(s + 8) * Dn + d, 0, 1);
            acc += sc[s] * xb[(size_t)s * Dn + d];
        }
        P[(size_t)(p * 64 + b) * Dn + d] = acc * inv;
    }
}

// ---------------------------------------------------------------------------
// Kernel D: feat = [sent_attn2 | sent_attn1 | out_per_seq[:,0,:] | entity2]
// ---------------------------------------------------------------------------
__global__ void feat_kernel(const float* __restrict__ x,
                            const float* __restrict__ E,
                            const float* __restrict__ P,
                            float* __restrict__ feat) {
    int b = blockIdx.x;
    float* fb = feat + (size_t)b * FD;
    for (int d = threadIdx.x; d < Dn; d += blockDim.x) {
        fb[d]            = P[(size_t)b * Dn + d];          // sent_attn2
        fb[Dn + d]       = P[(size_t)(64 + b) * Dn + d];   // sent_attn1
        fb[2 * Dn + d]   = x[(size_t)b * Sn * Dn + d];     // out_per_seq[:,0,:]
        fb[3 * Dn + d]   = E[(size_t)b * Dn + d];          // entity2
    }
}

// ---------------------------------------------------------------------------
// Kernel F: out[b] = <h[b,:], W2> + b2
// ---------------------------------------------------------------------------
__global__ void out_kernel(const float* __restrict__ h,
                           const float* __restrict__ W2,
                           const float* __restrict__ b2,
                           float* __restrict__ out) {
    __shared__ float red[8];
    int b = blockIdx.x;
    int tid = threadIdx.x, wave = tid >> 5, lane = tid & 31;
    float acc = 0.f;
    for (int n = tid; n < Hn; n += 256) acc += h[(size_t)b * Hn + n] * W2[n];
    for (int off = 16; off > 0; off >>= 1) acc += __shfl_xor(acc, off, 32);
    if (lane == 0) red[wave] = acc;
    __syncthreads();
    if (tid == 0) {
        float t = 0.f;
        for (int w = 0; w < 8; ++w) t += red[w];
        out[b] = t + b2[0];
    }
}

// ---------------------------------------------------------------------------
// Host launcher
// Inputs: 0 out_per_seq, 1 W_in, 2 W1, 3 b1, 4 alpha, 5 W2, 6 b2, 7 locs, 8 seps
// Workspace (floats): E[128*1536] | Q[128*1536] | P[128*1536] | feat[64*6144] | h[64*1024]
// ---------------------------------------------------------------------------
extern "C" void kernel_launch(void* const* d_in, const int* in_sizes, int n_in,
                              void* d_out, int out_size, void* d_ws, size_t ws_size,
                              hipStream_t stream) {
    const float* x     = (const float*)d_in[0];
    const float* W_in  = (const float*)d_in[1];
    const float* W1    = (const float*)d_in[2];
    const float* b1    = (const float*)d_in[3];
    const float* alpha = (const float*)d_in[4];
    const float* W2    = (const float*)d_in[5];
    const float* b2    = (const float*)d_in[6];
    const int*   locs  = (const int*)d_in[7];
    const int*   seps  = (const int*)d_in[8];
    float* out = (float*)d_out;

    float* ws   = (float*)d_ws;
    float* E    = ws;                         // 128*1536
    float* Q    = E + 128 * Dn;               // 128*1536
    float* P    = Q + 128 * Dn;               // 128*1536
    float* feat = P + 128 * Dn;               // 64*6144
    float* H    = feat + (size_t)Bn * FD;     // 64*1024

    // A: entity means
    entities_kernel<<<Bn, 256, 0, stream>>>(x, locs, E);

    // B: Q = E @ W_in^T   (M=128, N=1536, K=1536) -> 8x96 = 768 tiles, 4 waves/block
    wmma_gemm_kernel<false><<<768 / 4, 128, 0, stream>>>(
        E, W_in, Q, /*Mtiles=*/8, /*Ntiles=*/96, /*K=*/Dn, /*N=*/Dn, nullptr, nullptr);

    // C: attention pools (dominant HBM stream)
    attn_pool_kernel<<<dim3(Bn, 2), 256, 0, stream>>>(x, Q, locs, seps, P);

    // D: feature concat
    feat_kernel<<<Bn, 256, 0, stream>>>(x, E, P, feat);

    // E: H = leaky(feat @ W1^T + b1)  (M=64, N=1024, K=6144) -> 4x64 = 256 tiles
    wmma_gemm_kernel<true><<<256 / 4, 128, 0, stream>>>(
        feat, W1, H, /*Mtiles=*/4, /*Ntiles=*/64, /*K=*/FD, /*N=*/Hn, b1, alpha);

    // F: final projection
    out_kernel<<<Bn, 256, 0, stream>>>(H, W2, b2, out);
}